// Model_64845416235400
// MI455X (gfx1250) — compile-verified
//
#include <hip/hip_runtime.h>
#include <hip/hip_bf16.h>
#include <math.h>

typedef __attribute__((ext_vector_type(2))) float        v2f;
typedef __attribute__((ext_vector_type(8))) float        v8f;
typedef __attribute__((ext_vector_type(4))) unsigned int v4u;
typedef __attribute__((ext_vector_type(8))) int          v8i;
typedef __attribute__((ext_vector_type(4))) int          v4i;

#define B_      16
#define T_      4096
#define D_      512
#define C_      32
#define M_      64
#define NCHUNK  (T_ / C_)      /* 128 sequential steps */
#define NTHR    256            /* 8 wave32 per block   */

/* ---- LDS layout (in floats): double-buffered chunk for TDM pipeline ---- */
#define MEM_OFF    0
#define CHK0_OFF   (MEM_OFF   + M_ * D_)     /* 32768 */
#define CHK1_OFF   (CHK0_OFF  + C_ * D_)     /* 49152 */
#define LOG_OFF    (CHK1_OFF  + C_ * D_)     /* 65536 */
#define SCORE_OFF  (LOG_OFF   + C_ * M_)     /* 67584 */
#define MS_OFF     (SCORE_OFF + M_)
#define MF_OFF     (MS_OFF    + M_)
#define NEWM_OFF   (MF_OFF    + M_)
#define REDN_OFF   (NEWM_OFF  + D_)
#define REDD_OFF   (REDN_OFF  + 256)
#define REDNM_OFF  (REDD_OFF  + 256)
#define MISC_OFF   (REDNM_OFF + 4)
#define SMEM_FLOATS (MISC_OFF + 4)
#define SMEM_BYTES  (SMEM_FLOATS * 4)        /* ~269 KB, fits 320 KB/WGP */

#if __has_builtin(__builtin_amdgcn_tensor_load_to_lds) && \
    __has_builtin(__builtin_amdgcn_s_wait_tensorcnt)
#define HAVE_TDM 1
#else
#define HAVE_TDM 0
#endif

#if HAVE_TDM
/* Issue one TDM 2-D tile load: 32 rows x 512 f32, row stride 512, -> LDS.
   D# group0: count=1 | lds_addr | global_addr | type=2
   D# group1: data_size=4B; tensor_dim0=512, tensor_dim1=32;
              tile_dim0=512, tile_dim1=32, tile_dim2=0; dim0_stride=512.
   Groups 2/3 zero (dims 2..4 unused).                                    */
__device__ __forceinline__ void tdm_load_chunk(const float* gsrc, unsigned lds_byte_off)
{
    unsigned long long ga = (unsigned long long)(uintptr_t)gsrc;
    v4u g0;
    g0[0] = 1u;                                    /* count=1, user D#        */
    g0[1] = lds_byte_off;                          /* lds_addr (bytes)        */
    g0[2] = (unsigned)ga;                          /* global_addr[31:0]       */
    g0[3] = (unsigned)(ga >> 32) | (2u << 30);     /* global_addr[56:32]|type */
    v8i g1;
    g1[0] = (int)(2u << 16);                       /* mask=0, data_size=4B    */
    g1[1] = (int)((512u & 0xFFFFu) << 16);         /* tensor_dim0[15:0]       */
    g1[2] = (int)((512u >> 16) | (32u << 16));     /* td0[31:16] | td1[15:0]  */
    g1[3] = (int)((32u >> 16) | (512u << 16));     /* td1[31:16] | tile_dim0  */
    g1[4] = (int)32u;                              /* tile_dim1=32, tile_dim2=0 */
    g1[5] = (int)512u;                             /* dim0_stride[31:0]       */
    g1[6] = 0;                                     /* stride hi | dim1_stride */
    g1[7] = 0;
    v4i z4 = {0, 0, 0, 0};
    v8i z8 = {0, 0, 0, 0, 0, 0, 0, 0};
    /* 6-arg form (therock-10.0 headers / clang-23 lane) */
    __builtin_amdgcn_tensor_load_to_lds(g0, g1, z4, z4, z8, 0);
}
#endif

__global__ void ws_init_kernel(float* entArr, unsigned int* arrive) {
    int t = blockIdx.x * blockDim.x + threadIdx.x;
    if (t < NCHUNK * B_) entArr[t] = 0.0f;
    if (t == 0) *arrive = 0u;
}

__global__ __launch_bounds__(NTHR, 1)
void mem_bank_kernel(const float* __restrict__ x,
                     const float* __restrict__ mem_init,
                     float* __restrict__ out,
                     float* __restrict__ entArr,      /* [NCHUNK*B_] */
                     unsigned int* __restrict__ arrive)
{
    extern __shared__ float smem[];
    float* mem_s = smem + MEM_OFF;
    float* lg    = smem + LOG_OFF;
    float* scr   = smem + SCORE_OFF;
    float* msc   = smem + MS_OFF;
    float* mfr   = smem + MF_OFF;
    float* nwm   = smem + NEWM_OFF;
    float* redn  = smem + REDN_OFF;
    float* redd  = smem + REDD_OFF;
    float* rednm = smem + REDNM_OFF;
    int*   misc  = (int*)(smem + MISC_OFF);  /* [0]=branch flag, [1]=slot index */

    const int tid  = threadIdx.x;
    const int b    = blockIdx.x;
    const int wv   = tid >> 5;

#if HAVE_TDM
    const unsigned ldsbase = (unsigned)__builtin_amdgcn_groupstaticsize();
#endif

    /* init memory bank from mem_init; score/freq to zero */
    {
        const float4* mi   = (const float4*)(mem_init + (size_t)b * M_ * D_);
        float4*       mem4 = (float4*)mem_s;
        for (int i = tid; i < M_ * D_ / 4; i += NTHR) mem4[i] = mi[i];
        if (tid < M_) { msc[tid] = 0.0f; mfr[tid] = 0.0f; }
    }

#if HAVE_TDM
    /* pipeline prologue: DMA chunk 0 into buffer 0 (wave 0 only; EXEC-independent DMA) */
    if (wv == 0)
        tdm_load_chunk(x + (size_t)b * T_ * D_, ldsbase + CHK0_OFF * 4u);
#endif
    __syncthreads();

    const float inv_sqrt_d = 0.04419417382415922f;   /* 1/sqrt(512) */
    const float threshold  = 3.0f;                   /* 0.5*log2(64+1e-10) in f32 */

    for (int step = 0; step < NCHUNK; ++step) {
        const int    cur = step & 1;
        float* const chk = cur ? (smem + CHK1_OFF) : (smem + CHK0_OFF);

#if HAVE_TDM
        /* kick DMA for next chunk into the other buffer, then wait for current.
           TENSORcnt is per-wave and in-order: after issuing the next load there
           are 2 outstanding; waiting <=1 guarantees the current chunk landed.  */
        if (wv == 0) {
            if (step + 1 < NCHUNK) {
                const unsigned next_off = cur ? (ldsbase + CHK0_OFF * 4u)
                                              : (ldsbase + CHK1_OFF * 4u);
                tdm_load_chunk(x + ((size_t)b * T_ + (size_t)(step + 1) * C_) * D_,
                               next_off);
                __builtin_amdgcn_s_wait_tensorcnt(1);
            } else {
                __builtin_amdgcn_s_wait_tensorcnt(0);
            }
        }
#else
        {
            const float4* xg   = (const float4*)(x + ((size_t)b * T_ + (size_t)step * C_) * D_);
            float4*       chk4 = (float4*)chk;
            for (int i = tid; i < C_ * D_ / 4; i += NTHR) chk4[i] = xg[i];
            if (step + 1 < NCHUNK) {   /* global_prefetch_b8 for next chunk */
                const char* nx = (const char*)(x + ((size_t)b * T_ + (size_t)(step + 1) * C_) * D_);
                __builtin_prefetch(nx + (size_t)tid * 256, 0, 0);
            }
        }
#endif
        __syncthreads();

        /* ---- logits = chunk @ mem^T * inv_sqrt_d via V_WMMA_F32_16X16X4_F32 ----
           8 waves cover the 2x4 grid of 16x16 output tiles; K=512 in steps of 4. */
        {
            const int lane = tid & 31;
            const int mt   = wv >> 2;         /* M tile: 0..1 */
            const int nt   = wv & 3;          /* N tile: 0..3 */
            const int hi   = lane >> 4;       /* 0: K=0,1  1: K=2,3 */
            const int l    = lane & 15;
            const float* arow = chk   + (mt * 16 + l) * D_ + hi * 2;
            const float* brow = mem_s + (nt * 16 + l) * D_ + hi * 2;
            v8f acc = {};
            #pragma unroll 8
            for (int k = 0; k < D_; k += 4) {
                v2f a  = *(const v2f*)(arow + k);
                v2f bb = *(const v2f*)(brow + k);
                acc = __builtin_amdgcn_wmma_f32_16x16x4_f32(
                        false, a, false, bb, (short)0, acc, false, false);
            }
            /* C/D layout: VGPR r -> M = r (lanes 0-15), M = r+8 (lanes 16-31) */
            #pragma unroll
            for (int r = 0; r < 8; ++r)
                lg[(mt * 16 + hi * 8 + r) * M_ + nt * 16 + l] = acc[r] * inv_sqrt_d;
        }
        __syncthreads();

        /* ---- new_m = maxpool over chunk rows ---- */
        for (int d = tid; d < D_; d += NTHR) {
            float mx = chk[d];
            #pragma unroll 4
            for (int r = 1; r < C_; ++r) mx = fmaxf(mx, chk[r * D_ + d]);
            nwm[d] = mx;
        }
        /* ---- rowwise softmax over the 64 slots (in place) ---- */
        if (tid < C_) {
            float* row = lg + tid * M_;
            float mx = row[0];
            for (int m = 1; m < M_; ++m) mx = fmaxf(mx, row[m]);
            float s = 0.0f;
            for (int m = 0; m < M_; ++m) { float e = expf(row[m] - mx); row[m] = e; s += e; }
            float inv = 1.0f / s;
            for (int m = 0; m < M_; ++m) row[m] *= inv;
        }
        __syncthreads();
        /* ---- score = mean over chunk tokens ---- */
        if (tid < M_) {
            float s = 0.0f;
            for (int r = 0; r < C_; ++r) s += lg[r * M_ + tid];
            scr[tid] = s * (1.0f / C_);
        }
        __syncthreads();

        /* ---- batch-mean entropy: global arrival barrier over the 16 blocks ---- */
        if (tid == 0) {
            float ent = 0.0f;
            for (int m = 0; m < M_; ++m) { float s = scr[m]; ent -= log2f(s + 1e-10f) * s; }
            entArr[step * B_ + b] = ent;
            __threadfence();
            __hip_atomic_fetch_add(arrive, 1u, __ATOMIC_RELEASE, __HIP_MEMORY_SCOPE_AGENT);
            const unsigned int target = (unsigned int)(B_ * (step + 1));
            while (__hip_atomic_load(arrive, __ATOMIC_ACQUIRE, __HIP_MEMORY_SCOPE_AGENT) < target)
                __builtin_amdgcn_s_sleep(1);
            __threadfence();
            float es = 0.0f;                      /* fixed-order sum -> deterministic */
            for (int bb2 = 0; bb2 < B_; ++bb2) es += entArr[step * B_ + bb2];
            misc[0] = (es * (1.0f / B_) > threshold) ? 1 : 0;
        }
        __syncthreads();
        const bool take1 = (misc[0] != 0);

        if (take1) {
            /* path 1: replace lowest avg-score slot with new_m */
            if (tid == 0) {
                float best = 3.4e38f; int bi = 0;
                for (int m = 0; m < M_; ++m) {
                    float avg = (msc[m] + scr[m]) / (mfr[m] + 1.0f);
                    if (avg < best) { best = avg; bi = m; }
                }
                misc[1] = bi;
            }
            __syncthreads();
            const int repl = misc[1];
            for (int d = tid; d < D_; d += NTHR) mem_s[repl * D_ + d] = nwm[d];
            if (tid < M_) {
                if (tid == repl) { msc[tid] = 0.0f; mfr[tid] = 1.0f; }
                else             { msc[tid] += scr[tid]; mfr[tid] += 1.0f; }
            }
        } else {
            /* path 2: cosine-sim nearest slot merge (0.3*old + 0.7*new) */
            const int row = tid >> 2, part = tid & 3;
            const float* mrow = mem_s + row * D_ + part * 128;
            const float* nh   = nwm + part * 128;
            float n2 = 0.0f, dt = 0.0f;
            for (int i = 0; i < 128; ++i) {
                float mv = mrow[i], nv = nh[i];
                n2 += mv * mv; dt += mv * nv;
            }
            redn[tid] = n2; redd[tid] = dt;
            if (tid < 4) {
                float s = 0.0f;
                const float* nq = nwm + tid * 128;
                for (int i = 0; i < 128; ++i) s += nq[i] * nq[i];
                rednm[tid] = s;
            }
            __syncthreads();
            if (tid == 0) {
                float nm2  = rednm[0] + rednm[1] + rednm[2] + rednm[3];
                float ninv = 1.0f / sqrtf(nm2 + 1e-8f);
                float best = -3.4e38f; int bi = 0;
                for (int m = 0; m < M_; ++m) {
                    float r2 = redn[m*4] + redn[m*4+1] + redn[m*4+2] + redn[m*4+3];
                    float dd = redd[m*4] + redd[m*4+1] + redd[m*4+2] + redd[m*4+3];
                    float sim = dd * ninv / sqrtf(r2 + 1e-8f);
                    if (sim > best) { best = sim; bi = m; }
                }
                misc[1] = bi;
            }
            __syncthreads();
            const int mi2 = misc[1];
            for (int d = tid; d < D_; d += NTHR)
                mem_s[mi2 * D_ + d] = 0.3f * mem_s[mi2 * D_ + d] + 0.7f * nwm[d];
        }
        __syncthreads();
    }

    /* ---- write final memory bank ---- */
    {
        const float4* ms4 = (const float4*)mem_s;
        float4*       o4  = (float4*)(out + (size_t)b * M_ * D_);
        for (int i = tid; i < M_ * D_ / 4; i += NTHR) o4[i] = ms4[i];
    }
}

extern "C" void kernel_launch(void* const* d_in, const int* in_sizes, int n_in,
                              void* d_out, int out_size, void* d_ws, size_t ws_size,
                              hipStream_t stream) {
    (void)in_sizes; (void)n_in; (void)out_size; (void)ws_size;
    const float* x        = (const float*)d_in[0];
    const float* mem_init = (const float*)d_in[1];
    float*       out      = (float*)d_out;
    float*        entArr  = (float*)d_ws;
    unsigned int* arrive  = (unsigned int*)((char*)d_ws + (size_t)NCHUNK * B_ * sizeof(float));

    /* allow >64KB dynamic LDS (CDNA5 WGP has 320 KB) */
    (void)hipFuncSetAttribute((const void*)mem_bank_kernel,
                              hipFuncAttributeMaxDynamicSharedMemorySize, SMEM_BYTES);

    ws_init_kernel<<<dim3((NCHUNK * B_ + 255) / 256), dim3(256), 0, stream>>>(entArr, arrive);
    mem_bank_kernel<<<dim3(B_), dim3(NTHR), SMEM_BYTES, stream>>>(x, mem_init, out, entArr, arrive);
}